// OPTCacheFlowAttention_7206955123090
// MI455X (gfx1250) — compile-verified
//
#include <hip/hip_runtime.h>
#include <hip/hip_bf16.h>

typedef float v2f __attribute__((ext_vector_type(2)));
typedef float v8f __attribute__((ext_vector_type(8)));

// Shapes fixed by setup_inputs(): N=64, H=16, D=128, BS=16, X=8, MAXB=64 (T<=1024)
// key_cache  [NB, H, D/8, 16, 8]  -> (b*16+h)*2048 + (d>>3)*128 + pos*8 + (d&7)
// value_cache[NB, H, 16, 128]     -> (b*16+h)*2048 + pos*128 + d
//
// Split-KV flash-decode: 4 waves per (n,h), each owns tiles j = w, w+4, ...
// with a private online softmax; partials merged through LDS at the end.
#define NSPLIT 4

__global__ __launch_bounds__(32 * NSPLIT)
void paged_decode_attn(const float* __restrict__ query,
                       const float* __restrict__ kcache,
                       const float* __restrict__ vcache,
                       const int*   __restrict__ btab,
                       const int*   __restrict__ clens,
                       float* __restrict__ out)
{
    constexpr float SCALE = 0.08838834764831845f; // 1/sqrt(128)
    const int n    = blockIdx.x;    // token
    const int h    = blockIdx.y;    // head
    const int w    = threadIdx.y;   // wave id = context partition
    const int lane = threadIdx.x;   // wave32 lane
    const int pos  = lane & 15;     // position-in-block (A-matrix row M)
    const int half = lane >> 4;     // lane half selects K sub-pair (K+2,K+3)

    const int clen   = clens[n];
    const int ntiles = (clen + 15) >> 4;   // one 16-pos tile == one physical block

    // Preload q as B-matrix register pairs, broadcast across all 16 columns:
    // qp[i] = { q[4i + 2*half], q[4i + 1 + 2*half] }
    const float* qptr = query + ((size_t)(n * 16 + h)) * 128 + 2 * half;
    v2f qp[32];
#pragma unroll
    for (int i = 0; i < 32; ++i)
        qp[i] = *(const v2f*)(qptr + 4 * i);

    float m = -1e30f, lsum = 0.0f;
    float acc0 = 0.f, acc1 = 0.f, acc2 = 0.f, acc3 = 0.f; // out dims d0..d0+3
    const int d0   = lane << 2;
    const int koff = pos * 8 + 2 * half; // per-lane key-cache offset inside a k-step
    const int* btn = btab + n * 64;

    for (int j = w; j < ntiles; j += NSPLIT) {
        const int b = btn[j];                                  // uniform -> s_load
        const float* kb = kcache + ((size_t)(b * 16 + h)) * 2048;
        const float* vb = vcache + ((size_t)(b * 16 + h)) * 2048;

        // ---- issue ALL of this block's K+V traffic up front (16 KB in flight) --
        // A-matrix tiles: ka[i] lane<16 = {K[pos,4i],K[pos,4i+1]}, lane>=16 = {+2,+3}
        v2f ka[32];
#pragma unroll
        for (int i = 0; i < 32; ++i) {
            const int k0 = i << 2;
            ka[i] = *(const v2f*)(kb + (k0 >> 3) * 128 + (k0 & 7) + koff);
        }
        // V tile: coalesced b128 rows (512B = 32 lanes x 16B), consumed after softmax
        float4 vv[16];
#pragma unroll
        for (int t = 0; t < 16; ++t)
            vv[t] = *(const float4*)(vb + t * 128 + d0);

        // ---- logits tile: C[m_pos][*] = sum_d K[m_pos,d]*q[d], indep. chains ----
        v8f c0 = {}, c1 = {}, c2 = {}, c3 = {};
#pragma unroll
        for (int i = 0; i < 32; i += 4) {
            c0 = __builtin_amdgcn_wmma_f32_16x16x4_f32(false, ka[i+0], false, qp[i+0], (short)0, c0, false, false);
            c1 = __builtin_amdgcn_wmma_f32_16x16x4_f32(false, ka[i+1], false, qp[i+1], (short)0, c1, false, false);
            c2 = __builtin_amdgcn_wmma_f32_16x16x4_f32(false, ka[i+2], false, qp[i+2], (short)0, c2, false, false);
            c3 = __builtin_amdgcn_wmma_f32_16x16x4_f32(false, ka[i+3], false, qp[i+3], (short)0, c3, false, false);
        }
        const v8f c = c0 + c1 + c2 + c3;

        // C layout: vgpr r, lane half hh -> logits[r + 8*hh]; scale + length mask
        const int tbase = j * 16 + half * 8;
        float lg[8];
#pragma unroll
        for (int r = 0; r < 8; ++r) {
            const float x = c[r] * SCALE;
            lg[r] = (tbase + r < clen) ? x : -1e30f;
        }

        // online softmax: in-lane max over 8 positions + cross-half combine
        float tmax = lg[0];
#pragma unroll
        for (int r = 1; r < 8; ++r) tmax = fmaxf(tmax, lg[r]);
        tmax = fmaxf(tmax, __shfl_xor(tmax, 16, 32));
        const float mnew = fmaxf(m, tmax);
        const float corr = __expf(m - mnew);

        float p[8];
        float psum = 0.0f;
#pragma unroll
        for (int r = 0; r < 8; ++r) { p[r] = __expf(lg[r] - mnew); psum += p[r]; }
        psum += __shfl_xor(psum, 16, 32);

        lsum = lsum * corr + psum;
        acc0 *= corr; acc1 *= corr; acc2 *= corr; acc3 *= corr;
        m = mnew;

        // PV from registers: broadcast p[t] (lane 0: pos 0..7, lane 16: pos 8..15)
#pragma unroll
        for (int t = 0; t < 16; ++t) {
            const float pt = __shfl(p[t & 7], (t >> 3) << 4, 32);
            acc0 = fmaf(pt, vv[t].x, acc0);
            acc1 = fmaf(pt, vv[t].y, acc1);
            acc2 = fmaf(pt, vv[t].z, acc2);
            acc3 = fmaf(pt, vv[t].w, acc3);
        }
    }

    // ---- merge the NSPLIT wave-local partials through LDS -------------------
    __shared__ float s_acc[NSPLIT][128];
    __shared__ float s_m[NSPLIT];
    __shared__ float s_l[NSPLIT];

    *(float4*)&s_acc[w][d0] = make_float4(acc0, acc1, acc2, acc3);
    if (lane == 0) { s_m[w] = m; s_l[w] = lsum; }
    __syncthreads();

    if (w == 0) {
        float mstar = s_m[0];
#pragma unroll
        for (int p = 1; p < NSPLIT; ++p) mstar = fmaxf(mstar, s_m[p]);

        float lstar = 0.0f;
        float o0 = 0.f, o1 = 0.f, o2 = 0.f, o3 = 0.f;
#pragma unroll
        for (int p = 0; p < NSPLIT; ++p) {
            const float f = __expf(s_m[p] - mstar);  // 0 for waves with no tiles
            lstar += s_l[p] * f;
            const float4 a = *(const float4*)&s_acc[p][d0];
            o0 = fmaf(f, a.x, o0);
            o1 = fmaf(f, a.y, o1);
            o2 = fmaf(f, a.z, o2);
            o3 = fmaf(f, a.w, o3);
        }
        const float inv = 1.0f / lstar;
        *(float4*)(out + ((size_t)(n * 16 + h)) * 128 + d0) =
            make_float4(o0 * inv, o1 * inv, o2 * inv, o3 * inv);
    }
}

extern "C" void kernel_launch(void* const* d_in, const int* in_sizes, int n_in,
                              void* d_out, int out_size, void* d_ws, size_t ws_size,
                              hipStream_t stream) {
    const float* q  = (const float*)d_in[0];
    const float* kc = (const float*)d_in[1];
    const float* vc = (const float*)d_in[2];
    const int*   bt = (const int*)d_in[3];
    const int*   cl = (const int*)d_in[4];
    float* out = (float*)d_out;

    const int N = in_sizes[4];                  // 64 tokens
    const int H = in_sizes[0] / (N * 128);      // 16 heads

    dim3 grid(N, H), block(32, NSPLIT, 1);
    hipLaunchKernelGGL(paged_decode_attn, grid, block, 0, stream,
                       q, kc, vc, bt, cl, out);
}